// Residual_55989193670871
// MI455X (gfx1250) — compile-verified
//
#include <hip/hip_runtime.h>

typedef __attribute__((ext_vector_type(2))) float v2f;
typedef __attribute__((ext_vector_type(8))) float v8f;

#define N_NODES 50000
#define N_EDGES 800000
#define DD 128

// ---------------- utility kernels ----------------

__global__ void zero_kernel(float* __restrict__ p, int n) {
    int i = blockIdx.x * blockDim.x + threadIdx.x;
    int stride = gridDim.x * blockDim.x;
    for (; i < n; i += stride) p[i] = 0.0f;
}

__global__ void degree_kernel(const int* __restrict__ src, const int* __restrict__ dst,
                              float* __restrict__ deg_out, float* __restrict__ deg_in) {
    int e = blockIdx.x * blockDim.x + threadIdx.x;
    if (e < N_EDGES) {
        atomicAdd(&deg_out[src[e]], 1.0f);
        atomicAdd(&deg_in[dst[e]], 1.0f);
    }
}

__global__ void norm_kernel(float* __restrict__ a, float* __restrict__ b) {
    int i = blockIdx.x * blockDim.x + threadIdx.x;
    if (i < N_NODES) {
        a[i] = rsqrtf(fmaxf(a[i], 1.0f));
        b[i] = rsqrtf(fmaxf(b[i], 1.0f));
    }
}

// ---------------- GEMM1: h = (x * norm_src[:,None]) @ W_conv ----------------
// One wave32 per 16x16 output tile. Block = 256 thr = 8 waves = 16 rows x 128 cols.
// A (16x4 f32): lanes 0-15 hold K=k+0,k+1 ; lanes 16-31 hold K=k+2,k+3 (M = lane&15).
// B (4x16 f32): same striping with N = lane&15.
// C/D (16x16 f32): VGPR r -> M = r (lanes 0-15) / r+8 (lanes 16-31), N = lane&15.
__global__ void gemm1_wmma(const float* __restrict__ x, const float* __restrict__ norm_src,
                           const float* __restrict__ W, float* __restrict__ h) {
    const int lane   = threadIdx.x & 31;
    const int wave   = threadIdx.x >> 5;
    const int m_base = blockIdx.x * 16;
    const int n      = wave * 16 + (lane & 15);
    const int kk     = (lane >> 4) * 2;
    const int row    = m_base + (lane & 15);

    const float nv = norm_src[row];
    const float* __restrict__ xrow = x + (long)row * DD;

    v8f c = {};
#pragma unroll 4
    for (int k = 0; k < DD; k += 4) {
        v2f a, b;
        a.x = xrow[k + kk]     * nv;
        a.y = xrow[k + kk + 1] * nv;
        b.x = W[(k + kk)     * DD + n];
        b.y = W[(k + kk + 1) * DD + n];
        c = __builtin_amdgcn_wmma_f32_16x16x4_f32(false, a, false, b, (short)0, c, false, false);
    }

    const int rbase = m_base + ((lane >> 4) * 8);
#pragma unroll
    for (int r = 0; r < 8; ++r)
        h[(long)(rbase + r) * DD + n] = c[r];
}

// ---------------- edge scatter: agg[dst] += h[src] ----------------
// 1 thread per (edge, feature). h and agg are L2-resident (25.6MB each, 192MB L2).
__global__ void scatter_kernel(const int* __restrict__ src, const int* __restrict__ dst,
                               const float* __restrict__ h, float* __restrict__ agg) {
    long idx = (long)blockIdx.x * blockDim.x + threadIdx.x;
    int e = (int)(idx >> 7);
    int t = (int)(idx & (DD - 1));
    if (e < N_EDGES) {
        int s = src[e];
        int d = dst[e];
        atomicAdd(&agg[(long)d * DD + t], h[(long)s * DD + t]);
    }
}

// ---------------- GEMM2: out = concat(agg*norm_dst + b_conv, x) @ W_aggr + b_aggr ----
// K = 256 split in two uniform halves; conv normalization + bias and the skip
// concat are fused into the A-matrix load; b_aggr fused into the epilogue.
__global__ void gemm2_wmma(const float* __restrict__ agg, const float* __restrict__ norm_dst,
                           const float* __restrict__ feat, const float* __restrict__ b_conv,
                           const float* __restrict__ W, const float* __restrict__ b_aggr,
                           float* __restrict__ out) {
    const int lane   = threadIdx.x & 31;
    const int wave   = threadIdx.x >> 5;
    const int m_base = blockIdx.x * 16;
    const int n      = wave * 16 + (lane & 15);
    const int kk     = (lane >> 4) * 2;
    const int row    = m_base + (lane & 15);

    const float nd = norm_dst[row];
    const float* __restrict__ arow = agg  + (long)row * DD;
    const float* __restrict__ frow = feat + (long)row * DD;

    v8f c = {};
    // K = 0..127 : graph-conv branch of the concat
#pragma unroll 4
    for (int k = 0; k < DD; k += 4) {
        int kg = k + kk;
        v2f a, b;
        a.x = arow[kg]     * nd + b_conv[kg];
        a.y = arow[kg + 1] * nd + b_conv[kg + 1];
        b.x = W[kg       * DD + n];
        b.y = W[(kg + 1) * DD + n];
        c = __builtin_amdgcn_wmma_f32_16x16x4_f32(false, a, false, b, (short)0, c, false, false);
    }
    // K = 128..255 : skip-connection branch (raw features)
#pragma unroll 4
    for (int k = 0; k < DD; k += 4) {
        int kg = k + kk;
        v2f a, b;
        a.x = frow[kg];
        a.y = frow[kg + 1];
        b.x = W[(DD + kg)     * DD + n];
        b.y = W[(DD + kg + 1) * DD + n];
        c = __builtin_amdgcn_wmma_f32_16x16x4_f32(false, a, false, b, (short)0, c, false, false);
    }

    const float bn = b_aggr[n];
    const int rbase = m_base + ((lane >> 4) * 8);
#pragma unroll
    for (int r = 0; r < 8; ++r)
        out[(long)(rbase + r) * DD + n] = c[r] + bn;
}

// ---------------- launch ----------------

extern "C" void kernel_launch(void* const* d_in, const int* in_sizes, int n_in,
                              void* d_out, int out_size, void* d_ws, size_t ws_size,
                              hipStream_t stream) {
    (void)in_sizes; (void)n_in; (void)out_size; (void)ws_size;

    const float* feat   = (const float*)d_in[0];
    const int*   src    = (const int*)  d_in[1];
    const int*   dst    = (const int*)  d_in[2];
    const float* W_conv = (const float*)d_in[3];
    const float* b_conv = (const float*)d_in[4];
    const float* W_aggr = (const float*)d_in[5];
    const float* b_aggr = (const float*)d_in[6];
    float* out = (float*)d_out;

    float* ws       = (float*)d_ws;
    float* deg_out  = ws;                      // N  -> becomes norm_src in place
    float* deg_in   = ws + N_NODES;            // N  -> becomes norm_dst in place
    float* h        = ws + 2 * N_NODES;        // N*D
    float* agg      = h + (size_t)N_NODES * DD; // N*D

    // zero degrees (+contiguous) and agg accumulator
    zero_kernel<<<1024, 256, 0, stream>>>(deg_out, 2 * N_NODES);
    zero_kernel<<<8192, 256, 0, stream>>>(agg, N_NODES * DD);

    degree_kernel<<<(N_EDGES + 255) / 256, 256, 0, stream>>>(src, dst, deg_out, deg_in);
    norm_kernel<<<(N_NODES + 255) / 256, 256, 0, stream>>>(deg_out, deg_in);

    // GEMM1: 50000/16 = 3125 row tiles; 8 waves/block cover all 128 cols
    gemm1_wmma<<<N_NODES / 16, 256, 0, stream>>>(feat, deg_out, W_conv, h);

    // edge scatter: 800000 edges * 128 feats / 256 thr = 400000 blocks
    scatter_kernel<<<(long)N_EDGES * DD / 256, 256, 0, stream>>>(src, dst, h, agg);

    gemm2_wmma<<<N_NODES / 16, 256, 0, stream>>>(agg, deg_in, feat, b_conv,
                                                 W_aggr, b_aggr, out);
}